// GCN_Graph_30425548324935
// MI455X (gfx1250) — compile-verified
//
#include <hip/hip_runtime.h>
#include <math.h>

typedef float v2f __attribute__((ext_vector_type(2)));
typedef float v8f __attribute__((ext_vector_type(8)));

// ---------------------------------------------------------------------------
// GEMM: H[N,M] = X[N,K] @ W[K,M] (+bias, +relu), fp32 WMMA 16x16x4.
// One wave per 16x16 output tile; K,M compile-time so the K-loop fully
// unrolls into  global_load_b64 (A)  +  ds_load_b64 (B)  +  v_wmma.
// W is staged in LDS transposed with +4 dword pad: Wp[col*(K+4)+k], so each
// lane's (k, k+1) pair is one b64 load and the 32 lanes hit banks 4L..4L+3
// (conflict-free).  Requires N%16==0, M%16==0, K%4==0.
// ---------------------------------------------------------------------------
template <int K, int M>
__global__ void gemm_wmma_f32(const float* __restrict__ X,
                              const float* __restrict__ W,
                              float* __restrict__ H,
                              int N,
                              const float* __restrict__ bias,  // nullptr = none
                              int relu) {
  constexpr int KP = K + 4;
  __shared__ float Wp[KP * M];
  const int tid = threadIdx.x;
  // coalesced global read (k-major), transposed LDS write
  for (int i = tid; i < K * M; i += blockDim.x) {
    const int k = i / M;
    const int col = i - k * M;
    Wp[col * KP + k] = W[i];
  }
  __syncthreads();

  const int lane = tid & 31;
  const int wave = tid >> 5;
  constexpr int tilesN = M >> 4;
  const int totalTiles = (N >> 4) * tilesN;
  const int tile = blockIdx.x * (blockDim.x >> 5) + wave;
  if (tile >= totalTiles) return;  // wave-uniform: EXEC stays all-ones

  const int tm = tile / tilesN;
  const int tn = tile - tm * tilesN;
  const int row = (tm << 4) + (lane & 15);  // A: lanes 0-15 / 16-31 both M=0..15
  const int col = (tn << 4) + (lane & 15);  // B/C/D: column per lane%16
  const int khalf = (lane < 16) ? 0 : 2;    // upper half-wave holds the K+2 pair

  const float* xrow = X + (size_t)row * K + khalf;
  const float* wcol = &Wp[col * KP + khalf];
  v8f acc = {};
#pragma unroll
  for (int k0 = 0; k0 < K; k0 += 4) {
    v2f a = *(const v2f*)(xrow + k0);  // A[row][kk], A[row][kk+1]
    v2f b = *(const v2f*)(wcol + k0);  // W[kk][col], W[kk+1][col]
    acc = __builtin_amdgcn_wmma_f32_16x16x4_f32(false, a, false, b,
                                                (short)0, acc, false, false);
  }

  const float bv = bias ? bias[col] : 0.0f;
  const int rbase = (tm << 4) + ((lane < 16) ? 0 : 8);
#pragma unroll
  for (int r = 0; r < 8; ++r) {
    float v = acc[r] + bv;
    if (relu) v = fmaxf(v, 0.0f);
    H[(size_t)(rbase + r) * M + col] = v;  // D: VGPR r -> M=r / 8+r
  }
}

// ---------------------------------------------------------------------------
// Degree / normalization
// ---------------------------------------------------------------------------
__global__ void deg_scatter(const long long* __restrict__ dst,
                            float* __restrict__ deg, int E) {
  int e = blockIdx.x * blockDim.x + threadIdx.x;
  if (e < E) atomicAdd(deg + (int)dst[e], 1.0f);
}

__global__ void compute_dis(float* __restrict__ dis, int N) {
  int i = blockIdx.x * blockDim.x + threadIdx.x;
  if (i < N) dis[i] = rsqrtf(dis[i] + 1.0f);  // deg = in-degree + 1 (self loop)
}

// ---------------------------------------------------------------------------
// Edge scatter: AGG[dst] += H[src] * dis[src]*dis[dst]   (bandwidth-bound)
// ---------------------------------------------------------------------------
__global__ void scatter_edges(const float* __restrict__ H,
                              float* __restrict__ AGG,
                              const long long* __restrict__ src,
                              const long long* __restrict__ dst,
                              const float* __restrict__ dis,
                              int E, int M) {
  int e = blockIdx.x * blockDim.x + threadIdx.x;
  if (e >= E) return;
  const int s = (int)src[e];
  const int t = (int)dst[e];
  const float norm = dis[s] * dis[t];
  const float* hs = H + (size_t)s * M;
  float* at = AGG + (size_t)t * M;
  for (int d = 0; d < M; d += 4) {
    float4 v = *(const float4*)(hs + d);
    atomicAdd(at + d + 0, v.x * norm);
    atomicAdd(at + d + 1, v.y * norm);
    atomicAdd(at + d + 2, v.z * norm);
    atomicAdd(at + d + 3, v.w * norm);
  }
}

// OUT = relu(AGG + H * dis^2 + b)
__global__ void finalize_relu(const float* __restrict__ H,
                              const float* __restrict__ AGG,
                              const float* __restrict__ dis,
                              const float* __restrict__ bias,
                              float* __restrict__ OUT,
                              int N, int M) {
  int idx = blockIdx.x * blockDim.x + threadIdx.x;
  if (idx >= N * M) return;
  int i = idx / M;
  int d = idx - i * M;
  float s = dis[i];
  float v = AGG[idx] + H[idx] * s * s + bias[d];
  OUT[idx] = fmaxf(v, 0.0f);
}

// ---------------------------------------------------------------------------
// Global mean pool (feature dim fixed at 64)
// ---------------------------------------------------------------------------
__global__ void pool_scatter(const float* __restrict__ H,
                             const long long* __restrict__ batch,
                             float* __restrict__ sums,
                             float* __restrict__ cnts, int N) {
  int i = blockIdx.x * blockDim.x + threadIdx.x;
  if (i >= N) return;
  int g = (int)batch[i];
  const float* h = H + (size_t)i * 64;
  float* sg = sums + (size_t)g * 64;
  for (int d = 0; d < 64; d += 4) {
    float4 v = *(const float4*)(h + d);
    atomicAdd(sg + d + 0, v.x);
    atomicAdd(sg + d + 1, v.y);
    atomicAdd(sg + d + 2, v.z);
    atomicAdd(sg + d + 3, v.w);
  }
  atomicAdd(cnts + g, 1.0f);
}

__global__ void pool_div(const float* __restrict__ sums,
                         const float* __restrict__ cnts,
                         float* __restrict__ pooled, int total) {
  int idx = blockIdx.x * blockDim.x + threadIdx.x;
  if (idx >= total) return;
  int g = idx >> 6;
  pooled[idx] = sums[idx] / fmaxf(cnts[g], 1.0f);
}

// logits (bias already fused in GEMM) -> softmax rows of 16
__global__ void softmax16(const float* __restrict__ logits,
                          float* __restrict__ out) {
  __shared__ float l[16];
  __shared__ float p[16];
  int g = blockIdx.x;
  int j = threadIdx.x;  // blockDim == 16
  l[j] = logits[g * 16 + j];
  __syncthreads();
  float m = -1e30f;
  for (int k = 0; k < 16; ++k) m = fmaxf(m, l[k]);
  float e = expf(l[j] - m);
  p[j] = e;
  __syncthreads();
  float s = 0.0f;
  for (int k = 0; k < 16; ++k) s += p[k];
  out[g * 16 + j] = e / s;
}

// ---------------------------------------------------------------------------
// Host orchestration (stream-ordered, graph-capture safe)
// ---------------------------------------------------------------------------
extern "C" void kernel_launch(void* const* d_in, const int* in_sizes, int n_in,
                              void* d_out, int out_size, void* d_ws, size_t ws_size,
                              hipStream_t stream) {
  const float*     x    = (const float*)d_in[0];
  const long long* edge = (const long long*)d_in[1];
  const long long* bat  = (const long long*)d_in[2];
  const float* W1 = (const float*)d_in[3];  const float* b1 = (const float*)d_in[4];
  const float* W2 = (const float*)d_in[5];  const float* b2 = (const float*)d_in[6];
  const float* W3 = (const float*)d_in[7];  const float* b3 = (const float*)d_in[8];
  const float* fc1w = (const float*)d_in[9];  const float* fc1b = (const float*)d_in[10];
  const float* fc2w = (const float*)d_in[11]; const float* fc2b = (const float*)d_in[12];
  float* out = (float*)d_out;

  const int IN_DIM = 128;
  const int N = in_sizes[0] / IN_DIM;  // 100000 (divisible by 16)
  const int E = in_sizes[1] / 2;       // 3200000
  const int G = 512;
  const long long* src  = edge;
  const long long* dstp = edge + E;

  // workspace layout, 256B-aligned chunks
  float* w = (float*)d_ws;
  size_t off = 0;
  auto take = [&](size_t n) { float* p = w + off; off += (n + 63) & ~(size_t)63; return p; };
  float* dis    = take((size_t)N);
  float* bufH   = take((size_t)N * 64);
  float* bufA   = take((size_t)N * 64);
  float* bufO   = take((size_t)N * 64);
  float* sums   = take((size_t)G * 64);
  float* cnts   = take((size_t)G);
  float* pooled = take((size_t)G * 64);
  float* zbuf   = take((size_t)G * 32);
  float* lbuf   = take((size_t)G * 16);
  (void)ws_size; (void)n_in; (void)out_size;

  // degrees + symmetric norm
  hipMemsetAsync(dis, 0, (size_t)N * sizeof(float), stream);
  deg_scatter<<<(E + 255) / 256, 256, 0, stream>>>(dstp, dis, E);
  compute_dis<<<(N + 255) / 256, 256, 0, stream>>>(dis, N);

  // ---- GCN layer 1: 128 -> 32 ----
  {
    int tiles = (N / 16) * (32 / 16);
    gemm_wmma_f32<128, 32><<<(tiles + 3) / 4, 128, 0, stream>>>(x, W1, bufH, N, nullptr, 0);
    hipMemsetAsync(bufA, 0, (size_t)N * 32 * sizeof(float), stream);
    scatter_edges<<<(E + 255) / 256, 256, 0, stream>>>(bufH, bufA, src, dstp, dis, E, 32);
    finalize_relu<<<(N * 32 + 255) / 256, 256, 0, stream>>>(bufH, bufA, dis, b1, bufO, N, 32);
  }
  // ---- GCN layer 2: 32 -> 48 ----
  {
    int tiles = (N / 16) * (48 / 16);
    gemm_wmma_f32<32, 48><<<(tiles + 3) / 4, 128, 0, stream>>>(bufO, W2, bufH, N, nullptr, 0);
    hipMemsetAsync(bufA, 0, (size_t)N * 48 * sizeof(float), stream);
    scatter_edges<<<(E + 255) / 256, 256, 0, stream>>>(bufH, bufA, src, dstp, dis, E, 48);
    finalize_relu<<<(N * 48 + 255) / 256, 256, 0, stream>>>(bufH, bufA, dis, b2, bufO, N, 48);
  }
  // ---- GCN layer 3: 48 -> 64 ----
  {
    int tiles = (N / 16) * (64 / 16);
    gemm_wmma_f32<48, 64><<<(tiles + 3) / 4, 128, 0, stream>>>(bufO, W3, bufH, N, nullptr, 0);
    hipMemsetAsync(bufA, 0, (size_t)N * 64 * sizeof(float), stream);
    scatter_edges<<<(E + 255) / 256, 256, 0, stream>>>(bufH, bufA, src, dstp, dis, E, 64);
    finalize_relu<<<(N * 64 + 255) / 256, 256, 0, stream>>>(bufH, bufA, dis, b3, bufO, N, 64);
  }

  // global mean pool
  hipMemsetAsync(sums, 0, (size_t)G * 64 * sizeof(float), stream);
  hipMemsetAsync(cnts, 0, (size_t)G * sizeof(float), stream);
  pool_scatter<<<(N + 255) / 256, 256, 0, stream>>>(bufO, bat, sums, cnts, N);
  pool_div<<<(G * 64 + 255) / 256, 256, 0, stream>>>(sums, cnts, pooled, G * 64);

  // MLP head (WMMA with fused bias/relu) + softmax
  {
    int tiles = (G / 16) * (32 / 16);
    gemm_wmma_f32<64, 32><<<(tiles + 3) / 4, 128, 0, stream>>>(pooled, fc1w, zbuf, G, fc1b, 1);
  }
  {
    int tiles = (G / 16) * (16 / 16);
    gemm_wmma_f32<32, 16><<<(tiles + 3) / 4, 128, 0, stream>>>(zbuf, fc2w, lbuf, G, fc2b, 0);
  }
  softmax16<<<G, 16, 0, stream>>>(lbuf, out);
}